// ExtPointToNearestFaceDistance_9758165696603
// MI455X (gfx1250) — compile-verified
//
#include <hip/hip_runtime.h>
#include <hip/hip_bf16.h>

#define MIN_DIST_C 0.1f
#define BIG_C 1e10f
#define NCHUNKS 8            // face-dimension split per point tile

typedef __attribute__((ext_vector_type(2))) float v2f;
typedef __attribute__((ext_vector_type(8))) float v8f;

// ---------------------------------------------------------------------------
// Kernel 1: per-face precompute.
//   c    = MIN_DIST - dot(face_center, fn)
//   fnsq = dot(fn, fn)
// Stored SoA so WMMA B-tile loads are coalesced per 16-face column group.
// ---------------------------------------------------------------------------
__global__ void face_precompute_kernel(const float* __restrict__ meshV,
                                       const float* __restrict__ meshFN,
                                       const int*   __restrict__ meshF,
                                       float* __restrict__ fnx,
                                       float* __restrict__ fny,
                                       float* __restrict__ fnz,
                                       float* __restrict__ fc,
                                       float* __restrict__ fsq,
                                       int N, int F, int total) {
  int idx = blockIdx.x * blockDim.x + threadIdx.x;
  if (idx >= total) return;                 // total = B*F
  int b = idx / F;
  const int*   fi = meshF + (size_t)idx * 3;
  const float* Vb = meshV + (size_t)b * N * 3;

  float cx = 0.f, cy = 0.f, cz = 0.f;
#pragma unroll
  for (int k = 0; k < 3; ++k) {
    const float* v = Vb + (size_t)fi[k] * 3;
    cx += v[0]; cy += v[1]; cz += v[2];
  }
  const float third = 1.0f / 3.0f;
  cx *= third; cy *= third; cz *= third;

  const float* n = meshFN + (size_t)idx * 3;
  float nx = n[0], ny = n[1], nz = n[2];

  fnx[idx] = nx;
  fny[idx] = ny;
  fnz[idx] = nz;
  fc[idx]  = MIN_DIST_C - (cx * nx + cy * ny + cz * nz);
  fsq[idx] = nx * nx + ny * ny + nz * nz;
}

// ---------------------------------------------------------------------------
// Kernel 2: one wave per (16-point tile, face chunk).
// signed[p,f] via fp32 K=4 WMMA: A row = (px,py,pz,1), B col = (fnx,fny,fnz,c).
//
// A 16x4 f32 layout (ISA 7.12.2): lanes 0-15 hold K0/K1, lanes 16-31 K2/K3.
// B 4x16 f32 layout (symmetric):  VGPR0 = rows K0(l0-15)/K2(l16-31),
//                                 VGPR1 = rows K1/K3.
// C/D 16x16 f32: VGPR v, lane L -> M = v + 8*(L>>4), N = L&15.
//
// Tracks per-row running min(sqdist-or-BIG) and max(signed); all(signed<0)
// is recovered later as max(signed) < 0 (saves cmp+cndmask+or per element).
// ---------------------------------------------------------------------------
__global__ void __launch_bounds__(128)
point_face_min_kernel(const float* __restrict__ points,
                      const float* __restrict__ fnx,
                      const float* __restrict__ fny,
                      const float* __restrict__ fnz,
                      const float* __restrict__ fc,
                      const float* __restrict__ fsq,
                      float* __restrict__ pmin,   // [tile][chunk][16]
                      float* __restrict__ pmaxs,  // [tile][chunk][16]
                      int P, int F, int totalWaves, int ftPerChunk) {
  const int wavesPerBlock = blockDim.x >> 5;
  const int wave = threadIdx.x >> 5;
  const int lane = threadIdx.x & 31;
  const int gw   = blockIdx.x * wavesPerBlock + wave;
  if (gw >= totalWaves) return;            // wave-uniform: EXEC stays all-1s

  const int tile  = gw / NCHUNKS;          // global point tile (b * P/16 + pt)
  const int chunk = gw % NCHUNKS;
  const int tilesPerBatch = P >> 4;
  const int b  = tile / tilesPerBatch;
  const int pt = tile % tilesPerBatch;

  const int col  = lane & 15;              // N within tile (also M for A load)
  const int half = lane >> 4;

  // ---- A operand: point rows (loaded once per wave) ----
  const float* pp = points + ((size_t)b * P + (size_t)pt * 16 + col) * 3;
  float p0 = pp[0], p1 = pp[1], p2 = pp[2];
  v2f A;
  A.x = half ? p2 : p0;                    // K2 : K0
  A.y = half ? 1.0f : p1;                  // K3 : K1

  const size_t fbase = (size_t)b * F;

  float minv[8];
  float maxS[8];
#pragma unroll
  for (int v = 0; v < 8; ++v) { minv[v] = BIG_C; maxS[v] = -BIG_C; }

  const int ft0 = chunk * ftPerChunk;
  const int ft1 = ft0 + ftPerChunk;
#pragma unroll 2
  for (int ft = ft0; ft < ft1; ++ft) {
    const size_t face = fbase + (size_t)ft * 16 + col;
    const float q = fsq[face];
    v2f Bm;
    const float bx = fnx[face], by = fny[face], bz = fnz[face], bc = fc[face];
    Bm.x = half ? bz : bx;                 // K2 : K0
    Bm.y = half ? bc : by;                 // K3 : K1

    v8f acc = {0.f, 0.f, 0.f, 0.f, 0.f, 0.f, 0.f, 0.f};
    acc = __builtin_amdgcn_wmma_f32_16x16x4_f32(
        /*neg_a=*/false, A, /*neg_b=*/false, Bm,
        /*c_mod=*/(short)0, acc, /*reuse_a=*/false, /*reuse_b=*/false);

#pragma unroll
    for (int v = 0; v < 8; ++v) {
      float s = acc[v];
      maxS[v] = fmaxf(maxS[v], s);
      float sq = (s >= 0.0f) ? (s * s * q) : BIG_C;
      minv[v] = fminf(minv[v], sq);
    }
  }

  // ---- reduce the 16 face-columns (lanes within each half) ----
#pragma unroll
  for (int v = 0; v < 8; ++v) {
    float mv = minv[v];
    float ms = maxS[v];
#pragma unroll
    for (int off = 1; off < 16; off <<= 1) {
      mv = fminf(mv, __shfl_xor(mv, off, 32));
      ms = fmaxf(ms, __shfl_xor(ms, off, 32));
    }
    minv[v] = mv;
    maxS[v] = ms;
  }

  // ---- lanes 0 and 16 own rows M = v + 8*half; write chunk partials ----
  if (col == 0) {
    const size_t base = ((size_t)tile * NCHUNKS + chunk) * 16;
#pragma unroll
    for (int v = 0; v < 8; ++v) {
      pmin [base + v + 8 * half] = minv[v];
      pmaxs[base + v + 8 * half] = maxS[v];
    }
  }
}

// ---------------------------------------------------------------------------
// Kernel 3: combine chunks per point, apply exterior/inside rule, global mean.
// One thread per (b,p); wave-level sum then one atomicAdd per wave.
// ---------------------------------------------------------------------------
__global__ void reduce_kernel(const float* __restrict__ pmin,
                              const float* __restrict__ pmaxs,
                              const unsigned char* __restrict__ extflag,
                              float* __restrict__ out,
                              int totalPoints, float scale) {
  int idx = blockIdx.x * blockDim.x + threadIdx.x;
  float val = 0.f;
  if (idx < totalPoints) {
    const int tile = idx >> 4;             // = b*(P/16) + pt
    const int row  = idx & 15;
    float mv = BIG_C, ms = -BIG_C;
    const size_t base = (size_t)tile * NCHUNKS * 16 + row;
#pragma unroll
    for (int c = 0; c < NCHUNKS; ++c) {
      mv = fminf(mv, pmin [base + (size_t)c * 16]);
      ms = fmaxf(ms, pmaxs[base + (size_t)c * 16]);
    }
    bool inside = (!extflag[idx]) || (ms < 0.0f);  // interior OR all signed<0
    val = inside ? 0.0f : mv;
  }
#pragma unroll
  for (int off = 1; off < 32; off <<= 1)
    val += __shfl_xor(val, off, 32);
  if ((threadIdx.x & 31) == 0)
    atomicAdd(out, val * scale);
}

extern "C" void kernel_launch(void* const* d_in, const int* in_sizes, int n_in,
                              void* d_out, int out_size, void* d_ws, size_t ws_size,
                              hipStream_t stream) {
  const float* meshV  = (const float*)d_in[0];
  const float* points = (const float*)d_in[1];
  const float* meshFN = (const float*)d_in[2];
  const int*   meshF  = (const int*)d_in[3];
  const unsigned char* extflag = (const unsigned char*)d_in[4];
  float* out = (float*)d_out;

  const int B = 4;
  const int N = in_sizes[0] / (B * 3);
  const int P = in_sizes[1] / (B * 3);
  const int F = in_sizes[2] / (B * 3);

  // workspace layout:
  //   5 SoA face arrays of B*F floats, then 2 partial arrays of
  //   (B*P/16)*NCHUNKS*16 floats each.
  float* fnx = (float*)d_ws;
  float* fny = fnx + (size_t)B * F;
  float* fnz = fny + (size_t)B * F;
  float* fc  = fnz + (size_t)B * F;
  float* fsq = fc  + (size_t)B * F;
  const size_t tiles = (size_t)B * (P / 16);
  float* pmin  = fsq  + (size_t)B * F;
  float* pmaxs = pmin + tiles * NCHUNKS * 16;

  hipMemsetAsync(d_out, 0, sizeof(float) * out_size, stream);

  {
    int total = B * F;
    int threads = 256;
    int blocks = (total + threads - 1) / threads;
    face_precompute_kernel<<<blocks, threads, 0, stream>>>(
        meshV, meshFN, meshF, fnx, fny, fnz, fc, fsq, N, F, total);
  }

  {
    int totalWaves = (int)tiles * NCHUNKS;      // 4096 waves for B=4,P=2048
    int threads = 128;                          // 4 waves per block
    int wavesPerBlock = threads / 32;
    int blocks = (totalWaves + wavesPerBlock - 1) / wavesPerBlock;
    int ftPerChunk = (F / 16) / NCHUNKS;        // 64 face tiles per chunk
    point_face_min_kernel<<<blocks, threads, 0, stream>>>(
        points, fnx, fny, fnz, fc, fsq, pmin, pmaxs,
        P, F, totalWaves, ftPerChunk);
  }

  {
    int totalPoints = B * P;
    int threads = 256;
    int blocks = (totalPoints + threads - 1) / threads;
    float scale = 1.0f / (float)(B * P);
    reduce_kernel<<<blocks, threads, 0, stream>>>(
        pmin, pmaxs, extflag, out, totalPoints, scale);
  }
}